// AntiAliasInterpolation2d_7499012899331
// MI455X (gfx1250) — compile-verified
//
#include <hip/hip_runtime.h>
#include <math.h>

// Problem dims (fixed by the reference)
#define NIMG 32
#define IH   512
#define IW   512
#define CH   3
#define OH   128
#define OW   128
#define JCOLS (OW * CH)   // 384 flattened (ox, c) columns per image
#define KTAPS 13
#define KROWS 76          // 73 needed input rows, padded to 19 chunks * K=4
#define NCHUNK (KROWS / 4)
#define LSTRIDE 72        // LDS row stride (floats); rows 2 apart -> disjoint bank halves
#define JT 64             // (ox,c) columns per block
#define BANDROWS 16       // output rows per block
#define NBANDS (OH / BANDROWS)  // 8
#define NJT (JCOLS / JT)        // 6
#define ABIAS 60          // t = kk - 4*lm ranges over [-60, 74]; bias into [0, 134]
#define ATAB 144          // padded A-table length (float2 entries)

typedef float v2f __attribute__((ext_vector_type(2)));
typedef float v8f __attribute__((ext_vector_type(8)));

struct Args {
    float w[KTAPS];        // horizontal taps (constant-indexed only)
    float tab[2 * ATAB];   // interleaved pairs: tab[2i] = wext(i-60), tab[2i+1] = wext(i-59)
};

__global__ __launch_bounds__(128)
void aa_interp_fused(const float* __restrict__ x, float* __restrict__ out, Args a) {
    __shared__ float  Hs[KROWS * LSTRIDE];  // horizontally blurred rows, ~21.4 KB
    __shared__ float2 Atab[ATAB];           // zero-extended vertical tap pairs, 1.1 KB

    const int tid  = threadIdx.x;
    const int b    = blockIdx.x;
    const int n    = b / (NBANDS * NJT);
    const int rem  = b % (NBANDS * NJT);
    const int band = rem / NJT;
    const int jt   = rem % NJT;
    const int oy0  = band * BANDROWS;
    const int j0   = jt * JT;
    const int rows_base = 4 * oy0 - 6;      // input row for kk = 0

    // One-time A-table copy into LDS (dynamic kernarg reads are fine here).
    for (int i = tid; i < ATAB; i += 128)
        Atab[i] = make_float2(a.tab[2 * i], a.tab[2 * i + 1]);

    // ---- Pass 1: horizontal 13-tap blur at sampled columns only, into LDS ----
    // 76 rows x 64 cols = 4864 elements; 128 threads -> 38 iterations each (no tail).
    for (int e = tid; e < KROWS * JT; e += 128) {
        const int r    = e >> 6;            // LDS row (kk)
        const int jloc = e & 63;
        const int row  = rows_base + r;     // input image row
        float h = 0.0f;
        if (row >= 0 && row < IH) {
            const int j  = j0 + jloc;       // global flattened column
            const int ox = j / 3;
            const int c  = j - 3 * ox;
            const float* px = x + ((size_t)n * IH + row) * (size_t)(IW * CH) + c;
            const int xb = 4 * ox - 6;      // sampled center 4*ox, taps xb..xb+12
            if (xb >= 0 && xb + (KTAPS - 1) < IW) {
                // Interior: all 13 taps unguarded -> clean pipelined global loads.
                #pragma unroll
                for (int kx = 0; kx < KTAPS; ++kx)
                    h += a.w[kx] * px[(size_t)(xb + kx) * CH];
            } else {
                // Image edge (only first/last column tiles reach here).
                #pragma unroll
                for (int kx = 0; kx < KTAPS; ++kx) {
                    const int xi = xb + kx;
                    if (xi >= 0 && xi < IW) h += a.w[kx] * px[(size_t)xi * CH];
                }
            }
        }
        Hs[r * LSTRIDE + jloc] = h;         // out-of-image rows -> 0 (pad kk=73..75 get zero A weight)
    }
    __syncthreads();

    // ---- Pass 2: vertical 13-tap blur as D = A(16x76 banded weights) x B(76x16 data) ----
    const int wave  = tid >> 5;
    const int lane  = tid & 31;
    const int hi    = lane >> 4;            // half-wave select
    const int lm    = lane & 15;
    const int bcol  = wave * 16 + lm;       // B column for this lane
    const int tbase = 2 * hi - 4 * lm + ABIAS;  // A-table base index (per-lane constant)

    v8f acc = {};
    #pragma unroll
    for (int chunk = 0; chunk < NCHUNK; ++chunk) {  // 19 chunks of K=4
        // A[m, kk] = wext(kk - 4m): one unconditional 8B LDS load, no selects.
        const float2 ap = Atab[4 * chunk + tbase];
        v2f av; av[0] = ap.x; av[1] = ap.y;
        // B[kk, nn] = Hs[kk][bcol], kk = 4*chunk + 2*hi (+1 for second VGPR).
        const int kk = 4 * chunk + 2 * hi;
        v2f bv;
        bv[0] = Hs[kk * LSTRIDE + bcol];
        bv[1] = Hs[(kk + 1) * LSTRIDE + bcol];
#if __has_builtin(__builtin_amdgcn_wmma_f32_16x16x4_f32)
        acc = __builtin_amdgcn_wmma_f32_16x16x4_f32(
            /*neg_a=*/false, av, /*neg_b=*/false, bv,
            /*c_mod=*/(short)0, acc, /*reuse_a=*/false, /*reuse_b=*/false);
#else
        // Layout-exact scalar fallback: lane holds D rows m = v + 8*hi, col = bcol.
        #pragma unroll
        for (int v = 0; v < 8; ++v) {
            const int m = v + 8 * hi;
            #pragma unroll
            for (int kl = 0; kl < 4; ++kl) {
                const int kg = 4 * chunk + kl;
                const int t  = kg - 4 * m;
                if ((unsigned)t < (unsigned)KTAPS)
                    acc[v] += a.w[t] * Hs[kg * LSTRIDE + bcol];
            }
        }
#endif
    }

    // ---- Store D: VGPR v, lane -> (M = v + 8*hi, N = lane%16) ----
    const int j = j0 + bcol;
    #pragma unroll
    for (int v = 0; v < 8; ++v) {
        const int oy = oy0 + v + 8 * hi;
        out[((size_t)n * OH + oy) * (size_t)JCOLS + j] = acc[v];
    }
}

extern "C" void kernel_launch(void* const* d_in, const int* in_sizes, int n_in,
                              void* d_out, int out_size, void* d_ws, size_t ws_size,
                              hipStream_t stream) {
    (void)in_sizes; (void)n_in; (void)out_size; (void)d_ws; (void)ws_size;
    const float* x = (const float*)d_in[0];
    float* out = (float*)d_out;

    // Separable tap weights in double, rounded once to f32 (matches jnp fp32 pipeline
    // to within a ulp): w[t] = exp(-(t-6)^2 / (2*1.5^2)) / sum.
    double gd[KTAPS], s = 0.0;
    float wf[KTAPS];
    for (int i = 0; i < KTAPS; ++i) {
        const double d = (double)i - 6.0;
        gd[i] = exp(-(d * d) / 4.5);
        s += gd[i];
    }
    for (int i = 0; i < KTAPS; ++i) wf[i] = (float)(gd[i] / s);

    Args args;
    for (int i = 0; i < KTAPS; ++i) args.w[i] = wf[i];
    for (int i = 0; i < ATAB; ++i) {
        const int t0 = i - ABIAS;
        const int t1 = t0 + 1;
        args.tab[2 * i]     = (t0 >= 0 && t0 < KTAPS) ? wf[t0] : 0.0f;
        args.tab[2 * i + 1] = (t1 >= 0 && t1 < KTAPS) ? wf[t1] : 0.0f;
    }

    const dim3 grid(NIMG * NBANDS * NJT);   // 1536 blocks x 128 threads (4 waves)
    aa_interp_fused<<<grid, 128, 0, stream>>>(x, out, args);
}